// CameraAwareMemory_19765439496772
// MI455X (gfx1250) — compile-verified
//
#include <hip/hip_runtime.h>
#include <hip/hip_bf16.h>
#include <math.h>

typedef float v2f __attribute__((ext_vector_type(2)));
typedef float v4f __attribute__((ext_vector_type(4)));
typedef float v8f __attribute__((ext_vector_type(8)));

namespace {
constexpr int Bn = 256;     // batch
constexpr int Dk = 256;     // feature dim
constexpr int Pn = 32768;   // proxies
constexpr int Cn = 8;       // cameras
constexpr int BG = 50;      // BG_KNN
constexpr int PK = 3;       // POSK
constexpr float INVT = 20.0f;      // 1/TEMP
constexpr float BALW = 0.15f;      // BAL_W
constexpr int KC = 64;      // K chunk staged in LDS
constexpr int LPAD = 68;    // padded LDS row stride (floats): 272B rows, bank stride 4
}

// --- gfx1250 async global->LDS path (guarded; falls back to ld+ds_store) ---
#ifdef __has_builtin
#if __has_builtin(__builtin_amdgcn_global_load_async_to_lds_b128) && \
    __has_builtin(__builtin_amdgcn_s_wait_asynccnt)
#define ATHENA_ASYNC_LDS 1
#endif
#endif

#ifdef ATHENA_ASYNC_LDS
typedef int v4i_vs __attribute__((vector_size(4 * sizeof(int))));
typedef __attribute__((address_space(1))) v4i_vs* g_v4i_ptr;
typedef __attribute__((address_space(3))) v4i_vs* l_v4i_ptr;
#endif

__device__ __forceinline__ void async_copy16(const float* g, float* l) {
#ifdef ATHENA_ASYNC_LDS
  __builtin_amdgcn_global_load_async_to_lds_b128((g_v4i_ptr)g, (l_v4i_ptr)l, 0, 0);
#else
  *(v4f*)l = *(const v4f*)g;
#endif
}
__device__ __forceinline__ void async_join() {
#ifdef ATHENA_ASYNC_LDS
  __builtin_amdgcn_s_wait_asynccnt(0);
#endif
}

// order-preserving float -> uint key (larger float => larger key)
__device__ __forceinline__ unsigned fkey(float x) {
  unsigned u = __float_as_uint(x);
  return (u & 0x80000000u) ? ~u : (u | 0x80000000u);
}
__device__ __forceinline__ float fkeyinv(unsigned k) {
  unsigned u = (k & 0x80000000u) ? (k ^ 0x80000000u) : ~k;
  return __uint_as_float(u);
}

// ---------------------------------------------------------------------------
// prep: A_cat = [features ; mem[prx]]  (512 x 256), plus prx table
// ---------------------------------------------------------------------------
__global__ __launch_bounds__(256) void prep(const float* __restrict__ feat,
                                            const int* __restrict__ targets,
                                            const int* __restrict__ allProxy,
                                            const float* __restrict__ mem,
                                            float* __restrict__ Acat,
                                            int* __restrict__ prxBuf) {
  const int row = blockIdx.x;
  const int tid = threadIdx.x;
  if (row < Bn) {
    Acat[(size_t)row * Dk + tid] = feat[(size_t)row * Dk + tid];
  } else {
    const int r = row - Bn;
    const int prx = allProxy[targets[r]];
    if (tid == 0) prxBuf[r] = prx;
    Acat[(size_t)row * Dk + tid] = mem[(size_t)prx * Dk + tid];
  }
}

// ---------------------------------------------------------------------------
// gemm512: O[m][n] = sum_k Acat[m][k] * mem[n][k]   (M=512, N=32768, K=256)
// fp32 WMMA 16x16x4. Workgroup: 8 waves = 128 (M) x 64 (N) tile.
// A/B tiles staged in LDS via coalesced 16B copies (async-to-LDS when
// available); fragments fed from conflict-free ds_load_b64.
// ---------------------------------------------------------------------------
__global__ __launch_bounds__(256) void gemm512(const float* __restrict__ Acat,
                                               const float* __restrict__ mem,
                                               float* __restrict__ O) {
  __shared__ float Ash[128 * LPAD];  // 34 KB
  __shared__ float Bsh[64 * LPAD];   // 17 KB
  const int tid = threadIdx.x;
  const int lane = tid & 31;
  const int w = tid >> 5;        // wave id 0..7
  const int lo = lane & 15;      // M/N index within fragment
  const int hi = lane >> 4;      // K half select
  const int Mb0 = blockIdx.y * 128;
  const int Nb = blockIdx.x * 64;

  v8f acc0 = {}, acc1 = {}, acc2 = {}, acc3 = {};

  for (int kc = 0; kc < Dk; kc += KC) {
    // stage A tile: 128 rows x 64 cols = 2048 float4, 8 per thread (coalesced)
#pragma unroll
    for (int it = 0; it < 8; ++it) {
      const int i = tid + 256 * it;
      const int r = i >> 4;            // 16 float4 per row
      const int c = (i & 15) * 4;
      async_copy16(Acat + (size_t)(Mb0 + r) * Dk + kc + c, &Ash[r * LPAD + c]);
    }
    // stage B tile: 64 rows x 64 cols = 1024 float4, 4 per thread (coalesced)
#pragma unroll
    for (int it = 0; it < 4; ++it) {
      const int i = tid + 256 * it;
      const int r = i >> 4;
      const int c = (i & 15) * 4;
      async_copy16(mem + (size_t)(Nb + r) * Dk + kc + c, &Bsh[r * LPAD + c]);
    }
    async_join();
    __syncthreads();

    const float* arow = &Ash[(w * 16 + lo) * LPAD];
    const float* b0 = &Bsh[(0 * 16 + lo) * LPAD];
    const float* b1 = &Bsh[(1 * 16 + lo) * LPAD];
    const float* b2 = &Bsh[(2 * 16 + lo) * LPAD];
    const float* b3 = &Bsh[(3 * 16 + lo) * LPAD];
#pragma unroll
    for (int kb = 0; kb < KC; kb += 4) {
      const int kk = kb + 2 * hi;  // A 16x4: lanes 0-15 hold K=0,1; lanes 16-31 K=2,3
      v2f a  = *(const v2f*)(arow + kk);
      v2f f0 = *(const v2f*)(b0 + kk);
      v2f f1 = *(const v2f*)(b1 + kk);
      v2f f2 = *(const v2f*)(b2 + kk);
      v2f f3 = *(const v2f*)(b3 + kk);
      acc0 = __builtin_amdgcn_wmma_f32_16x16x4_f32(false, a, false, f0, (short)0, acc0, false, false);
      acc1 = __builtin_amdgcn_wmma_f32_16x16x4_f32(false, a, false, f1, (short)0, acc1, false, false);
      acc2 = __builtin_amdgcn_wmma_f32_16x16x4_f32(false, a, false, f2, (short)0, acc2, false, false);
      acc3 = __builtin_amdgcn_wmma_f32_16x16x4_f32(false, a, false, f3, (short)0, acc3, false, false);
    }
    __syncthreads();
  }

  // C/D layout: lane l, VGPR v -> row M = v + 8*(l>>4), col N = l&15
  const int Mb = Mb0 + w * 16;
#pragma unroll
  for (int v = 0; v < 8; ++v) {
    const size_t m = (size_t)(Mb + v + 8 * hi);
    O[m * Pn + (Nb + 0 * 16 + lo)] = acc0[v];
    O[m * Pn + (Nb + 1 * 16 + lo)] = acc1[v];
    O[m * Pn + (Nb + 2 * 16 + lo)] = acc2[v];
    O[m * Pn + (Nb + 3 * 16 + lo)] = acc3[v];
  }
}

// ---------------------------------------------------------------------------
// radix select: threshold key of the BG-th largest element (with exclusions)
// mode 0: value = inputs = r0*INVT, exclude the 8-wide positive block
// mode 1: value = sims = BALW*r0 + (1-BALW)*r1, exclude the 3 chosen indices
// Returns Tkey (key of BG-th largest) and rem = #ties at Tkey inside top-BG.
// ---------------------------------------------------------------------------
__device__ __forceinline__ bool excluded(int j, int mode, int posbase, const int* chosen) {
  if (mode == 0) return (j >= posbase) && (j < posbase + Cn);
  return (j == chosen[0]) || (j == chosen[1]) || (j == chosen[2]);
}
__device__ __forceinline__ float colval(const float* r0, const float* r1, int j, int mode) {
  const float a = r0[j];
  return (mode == 0) ? a * INVT : (BALW * a + (1.0f - BALW) * r1[j]);
}

__device__ void radix_select(const float* r0, const float* r1, int mode, int posbase,
                             const int* chosen, unsigned* hist,
                             unsigned* sh_prefix, int* sh_want,
                             unsigned& Tkey, int& rem) {
  const int tid = threadIdx.x;
  if (tid == 0) { *sh_prefix = 0u; *sh_want = BG; }
  __syncthreads();
  for (int round = 0; round < 4; ++round) {
    hist[tid] = 0u;
    __syncthreads();
    const unsigned pref = *sh_prefix;
    const int sh = 24 - 8 * round;
    for (int j = tid; j < Pn; j += 256) {
      if (excluded(j, mode, posbase, chosen)) continue;
      const unsigned u = fkey(colval(r0, r1, j, mode));
      if (round == 0 || (u >> (sh + 8)) == pref)
        atomicAdd(&hist[(u >> sh) & 255u], 1u);
    }
    __syncthreads();
    if (tid == 0) {
      const int want = *sh_want;
      int cum = 0, d = 255;
      for (; d >= 0; --d) {
        cum += (int)hist[d];
        if (cum >= want) break;
      }
      *sh_want = want - (cum - (int)hist[d]);
      *sh_prefix = (pref << 8) | (unsigned)d;
    }
    __syncthreads();
  }
  Tkey = *sh_prefix;
  rem = *sh_want;
  __syncthreads();
}

// ---------------------------------------------------------------------------
// rowreduce: one block per sample; produces intra/cross/online losses
// ---------------------------------------------------------------------------
__global__ __launch_bounds__(256) void rowreduce(const float* __restrict__ O,
                                                 const int* __restrict__ prxBuf,
                                                 const int* __restrict__ cams,
                                                 float* __restrict__ lossI,
                                                 float* __restrict__ lossC,
                                                 float* __restrict__ lossO) {
  const int b = blockIdx.x;
  const int tid = threadIdx.x;
  const float* r0 = O + (size_t)b * Pn;         // feature . mem  (raw score)
  const float* r1 = O + (size_t)(b + Bn) * Pn;  // mem[prx] . mem

  __shared__ float sval[256];
  __shared__ int sidx[256];
  __shared__ unsigned hist[256];
  __shared__ float camMax[Cn];
  __shared__ int camArg[Cn];
  __shared__ int chosen[PK];
  __shared__ float sh_rowmax, sh_maxcam;
  __shared__ unsigned sh_prefix;
  __shared__ int sh_want;
  __shared__ int tieCnt;
  __shared__ int tieIdx[128];

  const int prx = prxBuf[b];
  const int cam = cams[b];
  const int posbase = prx & ~(Cn - 1);  // pseudo-label block of 8 proxies

  // ---- scan 1: maxima + per-camera sims argmax (j%8 == tid%8 invariant) ----
  float lmaxin = -INFINITY, lmaxcam = -INFINITY, lsim = -INFINITY;
  int larg = 0x7fffffff;
  for (int j = tid; j < Pn; j += 256) {
    const float a = r0[j];
    const float in = a * INVT;
    const float sm = BALW * a + (1.0f - BALW) * r1[j];
    lmaxin = fmaxf(lmaxin, in);
    if ((j & (Cn - 1)) == cam) lmaxcam = fmaxf(lmaxcam, in);
    if (sm > lsim) { lsim = sm; larg = j; }  // strict > keeps lowest index
  }
  sval[tid] = lmaxin; __syncthreads();
  for (int s = 128; s > 0; s >>= 1) { if (tid < s) sval[tid] = fmaxf(sval[tid], sval[tid + s]); __syncthreads(); }
  if (tid == 0) sh_rowmax = sval[0];
  __syncthreads();
  sval[tid] = lmaxcam; __syncthreads();
  for (int s = 128; s > 0; s >>= 1) { if (tid < s) sval[tid] = fmaxf(sval[tid], sval[tid + s]); __syncthreads(); }
  if (tid == 0) sh_maxcam = sval[0];
  __syncthreads();
  sval[tid] = lsim; sidx[tid] = larg; __syncthreads();
  if (tid < Cn) {
    float bv = -INFINITY; int bi = 0x7fffffff;
    for (int t = tid; t < 256; t += Cn) {
      const float v = sval[t]; const int i = sidx[t];
      if (v > bv || (v == bv && i < bi)) { bv = v; bi = i; }
    }
    camMax[tid] = bv; camArg[tid] = bi;
  }
  __syncthreads();
  if (tid == 0) {  // top-3 cameras by max sim (ties -> lower camera index)
    unsigned used = 0;
    for (int r = 0; r < PK; ++r) {
      int best = 0; float bv = -INFINITY;
      for (int c2 = 0; c2 < Cn; ++c2) {
        if (used & (1u << c2)) continue;
        if (camMax[c2] > bv) { bv = camMax[c2]; best = c2; }
      }
      used |= 1u << best;
      chosen[r] = camArg[best];
    }
  }
  __syncthreads();

  // ---- intra: LSE over same-camera columns minus inputs[prx] ----
  {
    const float mc = sh_maxcam;
    float ls = 0.0f;
    for (int i = tid; i < Pn / Cn; i += 256) {
      const int j = cam + Cn * i;
      ls += expf(r0[j] * INVT - mc);
    }
    sval[tid] = ls; __syncthreads();
    for (int s = 128; s > 0; s >>= 1) { if (tid < s) sval[tid] += sval[tid + s]; __syncthreads(); }
    if (tid == 0) lossI[b] = mc + logf(sval[0]) - r0[prx] * INVT;
    __syncthreads();
  }

  // ---- cross: LSE over {8 positives U top-50 of inputs\pos} - mean(pos) ----
  {
    unsigned Tk; int rem;
    radix_select(r0, r1, 0, posbase, chosen, hist, &sh_prefix, &sh_want, Tk, rem);
    const float shift = sh_rowmax;
    float ls = 0.0f;
    for (int j = tid; j < Pn; j += 256) {
      if (j >= posbase && j < posbase + Cn) continue;
      const float in = r0[j] * INVT;
      if (fkey(in) > Tk) ls += expf(in - shift);
    }
    sval[tid] = ls; __syncthreads();
    for (int s = 128; s > 0; s >>= 1) { if (tid < s) sval[tid] += sval[tid + s]; __syncthreads(); }
    if (tid == 0) {
      float S = sval[0] + (float)rem * expf(fkeyinv(Tk) - shift);  // ties: identical values
      float posSum = 0.0f, posExp = 0.0f;
      for (int t = 0; t < Cn; ++t) {
        const float v = r0[posbase + t] * INVT;
        posSum += v; posExp += expf(v - shift);
      }
      lossC[b] = shift + logf(S + posExp) - posSum * (1.0f / Cn);
    }
    __syncthreads();
  }

  // ---- online: LSE over {3 chosen U top-50 of sims\chosen} - mean(chosen) ----
  {
    unsigned Tk; int rem;
    radix_select(r0, r1, 1, posbase, chosen, hist, &sh_prefix, &sh_want, Tk, rem);
    if (tid == 0) tieCnt = 0;
    __syncthreads();
    for (int j = tid; j < Pn; j += 256) {  // collect tie indices at threshold
      if (j == chosen[0] || j == chosen[1] || j == chosen[2]) continue;
      const float sm = BALW * r0[j] + (1.0f - BALW) * r1[j];
      if (fkey(sm) == Tk) { const int p = atomicAdd(&tieCnt, 1); if (p < 128) tieIdx[p] = j; }
    }
    __syncthreads();
    const float shift = sh_rowmax;
    float ls = 0.0f;
    for (int j = tid; j < Pn; j += 256) {
      if (j == chosen[0] || j == chosen[1] || j == chosen[2]) continue;
      const float sm = BALW * r0[j] + (1.0f - BALW) * r1[j];
      if (fkey(sm) > Tk) ls += expf(r0[j] * INVT - shift);
    }
    sval[tid] = ls; __syncthreads();
    for (int s = 128; s > 0; s >>= 1) { if (tid < s) sval[tid] += sval[tid + s]; __syncthreads(); }
    if (tid == 0) {
      const int tn = tieCnt < 128 ? tieCnt : 128;
      float tsum = 0.0f; int last = -1;
      for (int r = 0; r < rem; ++r) {  // jax top_k tie rule: lowest index first
        int bestj = 0x7fffffff;
        for (int t = 0; t < tn; ++t) { const int j = tieIdx[t]; if (j > last && j < bestj) bestj = j; }
        if (bestj == 0x7fffffff) break;
        last = bestj;
        tsum += expf(r0[bestj] * INVT - shift);
      }
      float chSum = 0.0f, chExp = 0.0f;
      for (int r = 0; r < PK; ++r) {
        const float v = r0[chosen[r]] * INVT;
        chSum += v; chExp += expf(v - shift);
      }
      lossO[b] = shift + logf(sval[0] + tsum + chExp) - chSum * (1.0f / PK);
    }
  }
}

// ---------------------------------------------------------------------------
// finalsum: camera-grouped mean of each loss, summed -> scalar
// ---------------------------------------------------------------------------
__global__ __launch_bounds__(256) void finalsum(const float* __restrict__ lossI,
                                                const float* __restrict__ lossC,
                                                const float* __restrict__ lossO,
                                                const int* __restrict__ cams,
                                                float* __restrict__ out) {
  __shared__ float ssum[3 * Cn];
  __shared__ int scnt[Cn];
  const int tid = threadIdx.x;
  if (tid < 3 * Cn) ssum[tid] = 0.0f;
  if (tid < Cn) scnt[tid] = 0;
  __syncthreads();
  if (tid < Bn) {
    const int c = cams[tid];
    atomicAdd(&ssum[c], lossI[tid]);
    atomicAdd(&ssum[Cn + c], lossC[tid]);
    atomicAdd(&ssum[2 * Cn + c], lossO[tid]);
    atomicAdd(&scnt[c], 1);
  }
  __syncthreads();
  if (tid == 0) {
    float tot = 0.0f;
    for (int c = 0; c < Cn; ++c) {
      if (scnt[c] > 0) {
        const float n = (float)scnt[c];
        tot += ssum[c] / n + ssum[Cn + c] / n + ssum[2 * Cn + c] / n;
      }
    }
    out[0] = tot;
  }
}

// ---------------------------------------------------------------------------
extern "C" void kernel_launch(void* const* d_in, const int* in_sizes, int n_in,
                              void* d_out, int out_size, void* d_ws, size_t ws_size,
                              hipStream_t stream) {
  const float* feat = (const float*)d_in[0];     // (B, D)
  const int* targets = (const int*)d_in[1];      // (B,)
  const int* cams = (const int*)d_in[2];         // (B,)
  // d_in[3] = epoch (unused)
  const float* mem = (const float*)d_in[4];      // (P, D)
  // d_in[5] = all_pseudo_label (derivable: proxy // C)
  const int* allProxy = (const int*)d_in[6];     // (N,)
  // d_in[7], d_in[8]: cam_proxies / label_proxies have closed-form structure

  float* ws = (float*)d_ws;
  float* O = ws;                                         // 512 * 32768 f32 (64 MB)
  float* Acat = ws + (size_t)(2 * Bn) * Pn;              // 512 * 256 f32
  int* prxBuf = (int*)(Acat + (size_t)(2 * Bn) * Dk);    // 256 ints
  float* lI = (float*)(prxBuf + Bn);
  float* lC = lI + Bn;
  float* lO = lC + Bn;

  prep<<<2 * Bn, 256, 0, stream>>>(feat, targets, allProxy, mem, Acat, prxBuf);

  dim3 g(Pn / 64, (2 * Bn) / 128);
  gemm512<<<g, 256, 0, stream>>>(Acat, mem, O);

  rowreduce<<<Bn, 256, 0, stream>>>(O, prxBuf, cams, lI, lC, lO);

  finalsum<<<1, 256, 0, stream>>>(lI, lC, lO, cams, (float*)d_out);
}